// GNN_8821862826461
// MI455X (gfx1250) — compile-verified
//
#include <hip/hip_runtime.h>
#include <hip/hip_bf16.h>

#define N_NODES  40000
#define N_EDGES  640000
#define N_GRAPHS 64

typedef __attribute__((ext_vector_type(2))) float v2f;
typedef __attribute__((ext_vector_type(8))) float v8f;

// ---------------------------------------------------------------------------
// Zero fill (graph-capture-safe, deterministic per replay)
// ---------------------------------------------------------------------------
__global__ void zero_f32(float* __restrict__ p, int n) {
    int i = blockIdx.x * blockDim.x + threadIdx.x;
    if (i < n) p[i] = 0.0f;
}

// ---------------------------------------------------------------------------
// WMMA f32 GEMM: out[nrows x dout] = act(A[nrows x din] @ W[din x dout]
//                                        + bias + Cadd)
// One wave32 per 16x16 output tile; K-loop of V_WMMA_F32_16X16X4_F32.
// ISA layouts (CDNA5 ISA §7.12.2):
//   A 16x4 : lane m = lane&15 is row; VGPR0/1 hold K = koff / koff+1,
//            koff = (lane>=16) ? 2 : 0.
//   B 4x16 : lane n = lane&15 is col; same K striping as A.
//   C/D    : VGPR v holds row = v + (lane>=16)*8, col = lane&15.
// All row/col dims here are multiples of 16; the only early-out branch is
// wave-uniform so EXEC is all-1s whenever WMMA executes.
// ---------------------------------------------------------------------------
template <int RELU, int HAS_BIAS, int HAS_ADD>
__global__ __launch_bounds__(256) void wmma_gemm_f32(
    const float* __restrict__ A, const float* __restrict__ W,
    const float* __restrict__ bias, const float* __restrict__ Cadd,
    float* __restrict__ out, int nrows, int din, int dout) {
    const int lane  = threadIdx.x & 31;
    const int wave  = blockIdx.x * (blockDim.x >> 5) + (threadIdx.x >> 5);
    const int tiles_n = dout >> 4;
    const int tile_m  = wave / tiles_n;
    const int tile_n  = wave - tile_m * tiles_n;
    if (tile_m * 16 >= nrows) return;           // wave-uniform guard

    const int m    = lane & 15;                 // A row / B,C col within tile
    const int half = lane >> 4;
    const int koff = half * 2;

    const float* Arow = A + (size_t)(tile_m * 16 + m) * din;
    const float* Wn   = W + tile_n * 16 + m;    // column n == m

    v8f acc = {};
    for (int kb = 0; kb < din; kb += 4) {
        v2f a, b;
        a.x = Arow[kb + koff];
        a.y = Arow[kb + koff + 1];
        b.x = Wn[(size_t)(kb + koff) * dout];
        b.y = Wn[(size_t)(kb + koff + 1) * dout];
        acc = __builtin_amdgcn_wmma_f32_16x16x4_f32(
            /*neg_a=*/false, a, /*neg_b=*/false, b,
            /*c_mod=*/(short)0, acc, /*reuse_a=*/false, /*reuse_b=*/false);
    }

    const int col = tile_n * 16 + m;
    float bv = 0.0f;
    if (HAS_BIAS) bv = bias[col];
    const int row0 = tile_m * 16 + half * 8;
#pragma unroll
    for (int v = 0; v < 8; ++v) {
        float val = acc[v] + bv;
        const size_t idx = (size_t)(row0 + v) * dout + col;
        if (HAS_ADD) val += Cadd[idx];
        if (RELU) val = fmaxf(val, 0.0f);
        out[idx] = val;
    }
}

// ---------------------------------------------------------------------------
// Edge gather + scatter-add: agg[dst[e]][:] += feat[src[e]][:]
// One thread per (edge, float4-chunk); chunk count is a power of two.
// ---------------------------------------------------------------------------
__global__ __launch_bounds__(256) void edge_scatter_add(
    const float* __restrict__ feat, const int* __restrict__ src,
    const int* __restrict__ dst, float* __restrict__ agg,
    int nedges, int dim, int chunk_shift) {
    const int t = blockIdx.x * blockDim.x + threadIdx.x;
    const int chunks = 1 << chunk_shift;
    if (t >= nedges * chunks) return;
    const int e = t >> chunk_shift;
    const int c = (t & (chunks - 1)) << 2;
    const int s = src[e];
    const int d = dst[e];
    const float4 v = *(const float4*)(feat + (size_t)s * dim + c);
    float* o = agg + (size_t)d * dim + c;
    atomicAdd(o + 0, v.x);
    atomicAdd(o + 1, v.y);
    atomicAdd(o + 2, v.z);
    atomicAdd(o + 3, v.w);
}

__global__ void relu_inplace(float* __restrict__ p, int n) {
    int i = blockIdx.x * blockDim.x + threadIdx.x;
    if (i < n) p[i] = fmaxf(p[i], 0.0f);
}

// ---------------------------------------------------------------------------
// Global mean pool accumulation: per-node atomics into [G x 16] sums + counts
// ---------------------------------------------------------------------------
__global__ __launch_bounds__(256) void pool_accum(
    const float* __restrict__ h, const int* __restrict__ batch,
    float* __restrict__ sums, float* __restrict__ counts, int nrows) {
    const int nidx = blockIdx.x * blockDim.x + threadIdx.x;
    if (nidx >= nrows) return;
    const int g = batch[nidx];
    const float* r = h + (size_t)nidx * 16;
#pragma unroll
    for (int f = 0; f < 16; ++f) atomicAdd(&sums[g * 16 + f], r[f]);
    atomicAdd(&counts[g], 1.0f);
}

__global__ void pool_fc(const float* __restrict__ sums,
                        const float* __restrict__ counts,
                        const float* __restrict__ Wfc,
                        const float* __restrict__ bfc,
                        float* __restrict__ out, int ngraphs) {
    const int g = blockIdx.x * blockDim.x + threadIdx.x;
    if (g >= ngraphs) return;
    const float inv = 1.0f / fmaxf(counts[g], 1.0f);
    float acc = 0.0f;
#pragma unroll
    for (int f = 0; f < 16; ++f) acc += sums[g * 16 + f] * inv * Wfc[f];
    out[g] = acc + bfc[0];
}

// ---------------------------------------------------------------------------
// Launch: exploit linearity of segment_sum to aggregate in min(din,dout) dim.
//   L1 (32->128): agg first (32-d), then GEMM with fused root/bias/relu.
//   L2..L4: GEMM first, scatter y_rel into (x@W_root + b), relu after (L2, L3 only).
// ---------------------------------------------------------------------------
extern "C" void kernel_launch(void* const* d_in, const int* in_sizes, int n_in,
                              void* d_out, int out_size, void* d_ws, size_t ws_size,
                              hipStream_t stream) {
    const float* x      = (const float*)d_in[0];
    const int*   edges  = (const int*)d_in[1];
    const int*   src    = edges;
    const int*   dst    = edges + N_EDGES;
    const int*   batch  = (const int*)d_in[2];
    const float* W1_rel = (const float*)d_in[3];
    const float* b1     = (const float*)d_in[4];
    const float* W1_root= (const float*)d_in[5];
    const float* W2_rel = (const float*)d_in[6];
    const float* b2     = (const float*)d_in[7];
    const float* W2_root= (const float*)d_in[8];
    const float* W3_rel = (const float*)d_in[9];
    const float* b3     = (const float*)d_in[10];
    const float* W3_root= (const float*)d_in[11];
    const float* W4_rel = (const float*)d_in[12];
    const float* b4     = (const float*)d_in[13];
    const float* W4_root= (const float*)d_in[14];
    const float* Wfc    = (const float*)d_in[15];
    const float* bfc    = (const float*)d_in[16];
    float* out = (float*)d_out;

    float* bufA   = (float*)d_ws;                       // N x 128
    float* bufB   = bufA + (size_t)N_NODES * 128;       // N x 128
    float* bufC   = bufB + (size_t)N_NODES * 128;       // N x 64 (agg / y_rel)
    float* sums   = bufC + (size_t)N_NODES * 64;        // G x 16
    float* counts = sums + N_GRAPHS * 16;               // G

    const int TB = 256;
    auto blk = [](long n, int t) { return (int)((n + t - 1) / t); };

    // ---- Layer 1: agg1 = seg_sum(x[src]) in 32-d; h1 = relu(agg1@W1_rel + b1 + x@W1_root)
    zero_f32<<<blk((long)N_NODES * 32, TB), TB, 0, stream>>>(bufC, N_NODES * 32);
    edge_scatter_add<<<blk((long)N_EDGES * 8, TB), TB, 0, stream>>>(
        x, src, dst, bufC, N_EDGES, 32, 3);
    wmma_gemm_f32<0, 0, 0><<<blk((long)(N_NODES / 16) * 8 * 32, TB), TB, 0, stream>>>(
        x, W1_root, nullptr, nullptr, bufA, N_NODES, 32, 128);          // t1
    wmma_gemm_f32<1, 1, 1><<<blk((long)(N_NODES / 16) * 8 * 32, TB), TB, 0, stream>>>(
        bufC, W1_rel, b1, bufA, bufB, N_NODES, 32, 128);                // h1 -> bufB

    // ---- Layer 2 (128->64): GEMM first, aggregate in 64-d
    wmma_gemm_f32<0, 0, 0><<<blk((long)(N_NODES / 16) * 4 * 32, TB), TB, 0, stream>>>(
        bufB, W2_rel, nullptr, nullptr, bufC, N_NODES, 128, 64);        // y_rel
    wmma_gemm_f32<0, 1, 0><<<blk((long)(N_NODES / 16) * 4 * 32, TB), TB, 0, stream>>>(
        bufB, W2_root, b2, nullptr, bufA, N_NODES, 128, 64);            // base
    edge_scatter_add<<<blk((long)N_EDGES * 16, TB), TB, 0, stream>>>(
        bufC, src, dst, bufA, N_EDGES, 64, 4);
    relu_inplace<<<blk((long)N_NODES * 64, TB), TB, 0, stream>>>(bufA, N_NODES * 64);

    // ---- Layer 3 (64->32)
    wmma_gemm_f32<0, 0, 0><<<blk((long)(N_NODES / 16) * 2 * 32, TB), TB, 0, stream>>>(
        bufA, W3_rel, nullptr, nullptr, bufC, N_NODES, 64, 32);
    wmma_gemm_f32<0, 1, 0><<<blk((long)(N_NODES / 16) * 2 * 32, TB), TB, 0, stream>>>(
        bufA, W3_root, b3, nullptr, bufB, N_NODES, 64, 32);
    edge_scatter_add<<<blk((long)N_EDGES * 8, TB), TB, 0, stream>>>(
        bufC, src, dst, bufB, N_EDGES, 32, 3);
    relu_inplace<<<blk((long)N_NODES * 32, TB), TB, 0, stream>>>(bufB, N_NODES * 32);

    // ---- Layer 4 (32->16), no ReLU
    wmma_gemm_f32<0, 0, 0><<<blk((long)(N_NODES / 16) * 1 * 32, TB), TB, 0, stream>>>(
        bufB, W4_rel, nullptr, nullptr, bufC, N_NODES, 32, 16);
    wmma_gemm_f32<0, 1, 0><<<blk((long)(N_NODES / 16) * 1 * 32, TB), TB, 0, stream>>>(
        bufB, W4_root, b4, nullptr, bufA, N_NODES, 32, 16);
    edge_scatter_add<<<blk((long)N_EDGES * 4, TB), TB, 0, stream>>>(
        bufC, src, dst, bufA, N_EDGES, 16, 2);

    // ---- Global mean pool + FC
    zero_f32<<<blk(N_GRAPHS * 17, TB), TB, 0, stream>>>(sums, N_GRAPHS * 17);
    pool_accum<<<blk(N_NODES, TB), TB, 0, stream>>>(bufA, batch, sums, counts, N_NODES);
    pool_fc<<<1, 64, 0, stream>>>(sums, counts, Wfc, bfc, out, N_GRAPHS);
}